// RNN_22256520528504
// MI455X (gfx1250) — compile-verified
//
#include <hip/hip_runtime.h>
#include <hip/hip_bf16.h>

// Problem constants (match the reference)
#define RB 64      // batch
#define RT 512     // timesteps
#define RD 512     // input dim
#define RH 1024    // hidden dim

typedef __attribute__((ext_vector_type(2))) float v2f;
typedef __attribute__((ext_vector_type(8))) float v8f;

#define LDS_PITCH 516   // 512 + 4 pad floats: rows hit distinct bank groups for ds_load_b64

// ---------------------------------------------------------------------------
// B-matrix pre-pack: fragment-native layout for V_WMMA_F32_16X16X4_F32.
// For k-tile kt (4 rows) and n-tile nt (16 cols), lane L holds
//   { M[4kt+kq][16nt + (L&15)], M[4kt+kq+1][...] },  kq = (L>>4)*2.
// Packed entry index = (kt*(RH/16) + nt)*32 + L  -> inner loop reads are one
// fully-coalesced global_load_b64 per fragment. M has leading dim RH.
// ---------------------------------------------------------------------------
__global__ void __launch_bounds__(256)
pack_b_kernel(const float* __restrict__ M, v2f* __restrict__ P, int K)
{
  const int tid   = blockIdx.x * 256 + threadIdx.x;
  const int total = (K / 4) * (RH / 16) * 32;
  if (tid >= total) return;
  const int lane = tid & 31;
  const int tile = tid >> 5;
  const int nt   = tile % (RH / 16);
  const int kt   = tile / (RH / 16);
  const int kq   = (lane >> 4) << 1;
  const int n    = nt * 16 + (lane & 15);
  const int k    = kt * 4 + kq;
  v2f v;
  v.x = M[(size_t)k * RH + n];
  v.y = M[(size_t)(k + 1) * RH + n];
  P[tid] = v;
}

// ---------------------------------------------------------------------------
// Shared GEMM core (macro keeps the __shared__ array in LDS address space so
// the compiler emits ds_* ops). Each wave computes a 16x32 strip: two 16x16
// f32 tiles (ACC0 at n-tile NT0, ACC1 at NT0+1) via V_WMMA_F32_16X16X4_F32,
// reusing each LDS A-fragment for both tiles. A is staged 16x512 per chunk.
// ---------------------------------------------------------------------------
#define GEMM_CORE2(ABASE, AROWSTRIDE, PACKB, KDIM, ACC0, ACC1, LANE, NT0)      \
  do {                                                                         \
    const int _m16 = (LANE) & 15;                                              \
    const int _kq  = ((LANE) >> 4) << 1;                                       \
    for (int _kc = 0; _kc < (KDIM); _kc += 512) {                              \
      /* cooperative stage of 16x512 f32 A-chunk into LDS (float4 loads) */    \
      _Pragma("unroll")                                                        \
      for (int _i = 0; _i < 8; ++_i) {                                         \
        int _f  = (int)threadIdx.x + _i * 256;   /* float4 index 0..2047 */    \
        int _r  = _f >> 7;                       /* row 0..15            */    \
        int _c4 = _f & 127;                      /* float4 col 0..127    */    \
        const float4 _v = *(const float4*)((ABASE) +                           \
                           (size_t)_r * (AROWSTRIDE) + _kc + _c4 * 4);         \
        *(float4*)&As[_r * LDS_PITCH + _c4 * 4] = _v;                          \
      }                                                                        \
      __syncthreads();                                                         \
      const v2f* _p = (PACKB) +                                                \
          ((size_t)(_kc >> 2) * (RH / 16) + (NT0)) * 32 + (LANE);              \
      _Pragma("unroll 4")                                                      \
      for (int _kb = 0; _kb < 512; _kb += 4) {                                 \
        v2f _a  = *(const v2f*)&As[_m16 * LDS_PITCH + _kb + _kq];              \
        v2f _b0 = _p[0];                                                       \
        v2f _b1 = _p[32];                                                      \
        (ACC0) = __builtin_amdgcn_wmma_f32_16x16x4_f32(                        \
                     false, _a, false, _b0, (short)0, (ACC0), false, false);   \
        (ACC1) = __builtin_amdgcn_wmma_f32_16x16x4_f32(                        \
                     false, _a, false, _b1, (short)0, (ACC1), false, false);   \
        _p += (RH / 16) * 32;                                                  \
      }                                                                        \
      __syncthreads();                                                         \
    }                                                                          \
  } while (0)

// ---------------------------------------------------------------------------
// Phase 1: xu[t*B+b][h] = sum_d x[b][t][d] * U[d][h] + bias[h]
// grid = (T*B/16, H/256), block = 256 (8 waves, 16x32 strip each)
// ---------------------------------------------------------------------------
__global__ void __launch_bounds__(256)
xu_gemm_kernel(const float* __restrict__ x, const v2f* __restrict__ packU,
               const float* __restrict__ bias, float* __restrict__ xu)
{
  __shared__ float As[16 * LDS_PITCH];

  const int lane = threadIdx.x & 31;
  const int wave = threadIdx.x >> 5;

  const int tb0 = blockIdx.x * 16;           // first flattened (t,b) row
  const int t   = tb0 / RB;
  const int b0  = tb0 % RB;
  const int nt0 = blockIdx.y * 16 + wave * 2;
  const int n0  = nt0 * 16 + (lane & 15);
  const int n1  = n0 + 16;

  const float*  Abase = x + (size_t)b0 * RT * RD + (size_t)t * RD;
  const size_t  Arow  = (size_t)RT * RD;     // stride between consecutive b

  v8f acc0 = {}, acc1 = {};
  GEMM_CORE2(Abase, Arow, packU, RD, acc0, acc1, lane, nt0);

  const float bv0 = bias[n0];
  const float bv1 = bias[n1];
  const int   mh  = (lane >> 4) << 3;        // +8 row offset for lanes 16..31
#pragma unroll
  for (int v = 0; v < 8; ++v) {
    const int m = v + mh;
    xu[(size_t)(tb0 + m) * RH + n0] = acc0[v] + bv0;
    xu[(size_t)(tb0 + m) * RH + n1] = acc1[v] + bv1;
  }
}

// ---------------------------------------------------------------------------
// Phase 2 (one launch per timestep, serial dependence through h):
//   hnext[b][h] = tanh(xu[t][b][h] + sum_k hprev[b][k] * W[k][h])
//   out[b][t][h] = hnext[b][h]   (written directly in [B,T,H] order)
// grid = (B/16, H/256), block = 256. Packed W (4 MB) stays L2-resident.
// ---------------------------------------------------------------------------
__global__ void __launch_bounds__(256)
rnn_step_kernel(const float* __restrict__ hprev, const v2f* __restrict__ packW,
                const float* __restrict__ xu, float* __restrict__ out,
                float* __restrict__ hnext, int t)
{
  __shared__ float As[16 * LDS_PITCH];

  const int lane = threadIdx.x & 31;
  const int wave = threadIdx.x >> 5;

  const int b0  = blockIdx.x * 16;
  const int nt0 = blockIdx.y * 16 + wave * 2;
  const int n0  = nt0 * 16 + (lane & 15);
  const int n1  = n0 + 16;

  const float* Abase = hprev + (size_t)b0 * RH;

  v8f acc0 = {}, acc1 = {};
  GEMM_CORE2(Abase, (size_t)RH, packW, RH, acc0, acc1, lane, nt0);

  const int mh = (lane >> 4) << 3;
#pragma unroll
  for (int v = 0; v < 8; ++v) {
    const int m = v + mh;
    const size_t xrow = ((size_t)t * RB + (b0 + m)) * RH;
    const float v0 = tanhf(acc0[v] + xu[xrow + n0]);
    const float v1 = tanhf(acc1[v] + xu[xrow + n1]);
    const size_t orow = (size_t)(b0 + m) * RT * RH + (size_t)t * RH;
    out[orow + n0] = v0;
    out[orow + n1] = v1;
    const size_t hrow = (size_t)(b0 + m) * RH;
    hnext[hrow + n0] = v0;
    hnext[hrow + n1] = v1;
  }
}

// ---------------------------------------------------------------------------
// Launch: pack U/W, phase-1 GEMM, copy initial state, 512 dependent step
// kernels (ping-pong h buffers in workspace), copy h_last to output tail.
// ---------------------------------------------------------------------------
extern "C" void kernel_launch(void* const* d_in, const int* in_sizes, int n_in,
                              void* d_out, int out_size, void* d_ws, size_t ws_size,
                              hipStream_t stream) {
  (void)in_sizes; (void)n_in; (void)out_size; (void)ws_size;

  const float* x     = (const float*)d_in[0];   // [B,T,D]
  const float* state = (const float*)d_in[1];   // [B,H]
  const float* U     = (const float*)d_in[2];   // [D,H]
  const float* W     = (const float*)d_in[3];   // [H,H]
  const float* bias  = (const float*)d_in[4];   // [H]
  float*       out   = (float*)d_out;           // [B,T,H] ++ [B,H]

  char*  ws = (char*)d_ws;
  size_t off = 0;
  float* xu = (float*)(ws + off);  off += (size_t)RT * RB * RH * sizeof(float); // 128 MB
  float* h0 = (float*)(ws + off);  off += (size_t)RB * RH * sizeof(float);
  float* h1 = (float*)(ws + off);  off += (size_t)RB * RH * sizeof(float);
  v2f* packU = (v2f*)(ws + off);   off += (size_t)(RD / 4) * (RH / 16) * 32 * sizeof(v2f); // 2 MB
  v2f* packW = (v2f*)(ws + off);   off += (size_t)(RH / 4) * (RH / 16) * 32 * sizeof(v2f); // 4 MB

  // One-shot fragment packing of U and W.
  {
    const int totU = (RD / 4) * (RH / 16) * 32;
    const int totW = (RH / 4) * (RH / 16) * 32;
    pack_b_kernel<<<(totU + 255) / 256, 256, 0, stream>>>(U, packU, RD);
    pack_b_kernel<<<(totW + 255) / 256, 256, 0, stream>>>(W, packW, RH);
  }

  // Phase 1: xu = einsum('btd,dh->tbh', x, U) + b
  dim3 g1((RT * RB) / 16, RH / 256);
  xu_gemm_kernel<<<g1, 256, 0, stream>>>(x, packU, bias, xu);

  // h0 <- initial state
  hipMemcpyAsync(h0, state, (size_t)RB * RH * sizeof(float),
                 hipMemcpyDeviceToDevice, stream);

  // Phase 2: sequential scan, one launch per step (graph replay amortizes)
  dim3 g2(RB / 16, RH / 256);
  float* hprev = h0;
  float* hnext = h1;
  for (int t = 0; t < RT; ++t) {
    rnn_step_kernel<<<g2, 256, 0, stream>>>(hprev, packW, xu, out, hnext, t);
    float* tmp = hprev; hprev = hnext; hnext = tmp;
  }

  // h_last (result of step T-1) -> output tail
  hipMemcpyAsync(out + (size_t)RB * RT * RH, hprev,
                 (size_t)RB * RH * sizeof(float),
                 hipMemcpyDeviceToDevice, stream);
}